// GMMILDiscriminator_1606317769362
// MI455X (gfx1250) — compile-verified
//
#include <hip/hip_runtime.h>
#include <hip/hip_bf16.h>

// ---------------------------------------------------------------------------
// GMMIL discriminator, fused for gfx1250 (MI455X).
//   N1 = N2 = 4096, feature dim = 48 + 16 = 64.
//   Distance tiles computed with chained V_WMMA_F32_16X16X4_F32 (exact fp32).
//   Weighted medians via two-level device-side weighted-histogram quantile.
// ---------------------------------------------------------------------------

#define N       4096
#define DFEAT   64
#define NSTATE  48
#define NACT    16
#define NB      8192          // histogram bins
#define HI0     8.0f          // coarse histogram range [0, HI0)

typedef float v2f __attribute__((ext_vector_type(2)));
typedef float v8f __attribute__((ext_vector_type(8)));

// Workspace layout (floats)
#define WS_SA    0                         // [4096 x 64]
#define WS_ESA   (WS_SA + N * DFEAT)       // [4096 x 64]
#define WS_X2    (WS_ESA + N * DFEAT)      // [4096]
#define WS_E2    (WS_X2 + N)               // [4096]
#define WS_HIST  (WS_E2 + N)               // [8192]
#define WS_P     (WS_HIST + NB)            // [16]: 0=lo 1=scale 2=g1 3=g2 4=Sw 5=Sew
#define WS_TOTAL (WS_P + 16)

// ---------------------------------------------------------------------------
// 16x16 fp32 tile of X(16x64) * Y(16x64)^T via 16 chained WMMA f32 16x16x4.
// A lane layout (16x4, MxK): lane L: m = L&15; VGPR0 = K0+2*(L>>4), VGPR1 = +1
// B lane layout (4x16, KxN): lane L: n = L&15; same K striping (B = Y^T so
// B[k][n] = Y[n][k] -> identical per-lane float2 load pattern as A).
// C/D layout: lane L: n = L&15; c[r] is row m = r + 8*(L>>4).
// ---------------------------------------------------------------------------
__device__ __forceinline__ v8f wmma_xyT_tile(const float* __restrict__ X,
                                             const float* __restrict__ Y,
                                             int i0, int j0) {
    const int lane = threadIdx.x & 31;
    const int row  = lane & 15;
    const int kh   = (lane >> 4) & 1;          // 0 -> k,k+1 ; 1 -> k+2,k+3
    const float* xr = X + (size_t)(i0 + row) * DFEAT + 2 * kh;
    const float* yr = Y + (size_t)(j0 + row) * DFEAT + 2 * kh;
    v8f c = {};
#pragma unroll
    for (int k0 = 0; k0 < DFEAT; k0 += 4) {
        v2f a = *(const v2f*)(xr + k0);
        v2f b = *(const v2f*)(yr + k0);
        c = __builtin_amdgcn_wmma_f32_16x16x4_f32(
                /*neg_a=*/false, a, /*neg_b=*/false, b,
                /*c_mod=*/(short)0, c, /*reuse_a=*/false, /*reuse_b=*/false);
    }
    return c;
}

// ---------------------------------------------------------------------------
// Zero histogram + params; serial (deterministic) weight sums.
// ---------------------------------------------------------------------------
__global__ void k_zero(const float* __restrict__ w, const float* __restrict__ ew,
                       float* __restrict__ hist, float* __restrict__ P) {
    if (blockIdx.x == 0) {
        if (threadIdx.x == 0) {
            float s = 0.f;
            for (int i = 0; i < N; ++i) s += w[i];
            P[4] = s;
            P[0] = 0.f;
            P[1] = (float)NB / HI0;
            P[2] = 0.f;
        }
        if (threadIdx.x == 1) {
            float s = 0.f;
            for (int i = 0; i < N; ++i) s += ew[i];
            P[5] = s;
            P[3] = 0.f;
        }
    }
    for (int b = blockIdx.x * blockDim.x + threadIdx.x; b < NB;
         b += gridDim.x * blockDim.x)
        hist[b] = 0.f;
}

// ---------------------------------------------------------------------------
// Concatenate [state|action] rows into 64-wide matrices, compute row norms.
// ---------------------------------------------------------------------------
__global__ void k_prep(const float* __restrict__ state, const float* __restrict__ action,
                       const float* __restrict__ estate, const float* __restrict__ eaction,
                       float* __restrict__ SA, float* __restrict__ ESA,
                       float* __restrict__ X2, float* __restrict__ E2) {
    int i = blockIdx.x * blockDim.x + threadIdx.x;
    if (i >= N) return;
    float s = 0.f;
    for (int k = 0; k < NSTATE; ++k) {
        float v = state[(size_t)i * NSTATE + k];
        SA[(size_t)i * DFEAT + k] = v; s += v * v;
    }
    for (int k = 0; k < NACT; ++k) {
        float v = action[(size_t)i * NACT + k];
        SA[(size_t)i * DFEAT + NSTATE + k] = v; s += v * v;
    }
    X2[i] = s;
    s = 0.f;
    for (int k = 0; k < NSTATE; ++k) {
        float v = estate[(size_t)i * NSTATE + k];
        ESA[(size_t)i * DFEAT + k] = v; s += v * v;
    }
    for (int k = 0; k < NACT; ++k) {
        float v = eaction[(size_t)i * NACT + k];
        ESA[(size_t)i * DFEAT + NSTATE + k] = v; s += v * v;
    }
    E2[i] = s;
}

// ---------------------------------------------------------------------------
// Weighted histogram of D = max(x2+y2-2*X.Y^T, 0)/64 with weight wx_i*wy_j.
// One block = one 16-row strip; 8 waves split the 256 column tiles.
// ---------------------------------------------------------------------------
__global__ void __launch_bounds__(256)
k_hist(const float* __restrict__ X, const float* __restrict__ Y,
       const float* __restrict__ x2, const float* __restrict__ y2,
       const float* __restrict__ wx, const float* __restrict__ wy,
       const float* __restrict__ P, float* __restrict__ hist) {
    __shared__ float lh[NB];
    for (int b = threadIdx.x; b < NB; b += blockDim.x) lh[b] = 0.f;
    __syncthreads();

    const float lo    = P[0];
    const float scale = P[1];
    const int i0   = blockIdx.x * 16;
    const int wave = threadIdx.x >> 5;
    const int lane = threadIdx.x & 31;
    const int nrow = lane & 15;
    const int half = lane >> 4;

    float xm[8], wxm[8];
#pragma unroll
    for (int r = 0; r < 8; ++r) {
        int m = r + 8 * half;
        xm[r]  = x2[i0 + m];
        wxm[r] = wx[i0 + m];
    }

    for (int jt = wave; jt < N / 16; jt += 8) {
        const int j0 = jt * 16;
        v8f c = wmma_xyT_tile(X, Y, i0, j0);
        const float yn  = y2[j0 + nrow];
        const float wyn = wy[j0 + nrow];
#pragma unroll
        for (int r = 0; r < 8; ++r) {
            float d2 = fmaxf(xm[r] + yn - 2.0f * c[r], 0.f) * (1.0f / DFEAT);
            int b = (int)((d2 - lo) * scale);
            b = min(max(b, 0), NB - 1);
            atomicAdd(&lh[b], wxm[r] * wyn);
        }
    }
    __syncthreads();
    for (int b = threadIdx.x; b < NB; b += blockDim.x)
        if (lh[b] != 0.f) atomicAdd(&hist[b], lh[b]);
}

// ---------------------------------------------------------------------------
// Locate weighted 0.5-quantile bin.
//  mode 0: refine -> shrink [lo, lo+1/scale) range by NB, zero hist.
//  mode 1: finalize -> P[gslot] = 1/(median + 1e-8), reset range, zero hist.
// ---------------------------------------------------------------------------
__global__ void k_select(float* __restrict__ hist, float* __restrict__ P,
                         int mode, int gslot) {
    if (threadIdx.x == 0) {
        const float target = 0.5f * P[4] * P[5];
        float cum = 0.f;
        int b = NB - 1;
        for (int i = 0; i < NB; ++i) {
            cum += hist[i];
            if (cum >= target) { b = i; break; }
        }
        const float lo = P[0], scale = P[1];
        if (mode == 0) {
            P[0] = lo + (float)b / scale;   // new lower edge
            P[1] = scale * (float)NB;       // bin width shrinks by NB
        } else {
            const float med = lo + ((float)b + 0.5f) / scale;
            P[gslot] = 1.0f / (med + 1e-8f);
            P[0] = 0.f;                     // reset for next coarse pass
            P[1] = (float)NB / HI0;
        }
    }
    __syncthreads();
    for (int i = threadIdx.x; i < NB; i += blockDim.x) hist[i] = 0.f;
}

// ---------------------------------------------------------------------------
// out[i] = wn_i * ( (K_ae @ ewn)[i] - (K_aa @ wn)[i] ),  K = e^{-g1 D}+e^{-g2 D}
// One block per 16-row strip; deterministic shuffle + LDS reductions.
// ---------------------------------------------------------------------------
__global__ void __launch_bounds__(256)
k_out(const float* __restrict__ SA, const float* __restrict__ ESA,
      const float* __restrict__ X2, const float* __restrict__ E2,
      const float* __restrict__ w, const float* __restrict__ ew,
      const float* __restrict__ P, float* __restrict__ out) {
    __shared__ float part_ae[8][16];
    __shared__ float part_aa[8][16];

    const float g1 = P[2], g2 = P[3], Sw = P[4], Sew = P[5];
    const int i0   = blockIdx.x * 16;
    const int wave = threadIdx.x >> 5;
    const int lane = threadIdx.x & 31;
    const int nrow = lane & 15;
    const int half = lane >> 4;

    float xm[8];
#pragma unroll
    for (int r = 0; r < 8; ++r) xm[r] = X2[i0 + r + 8 * half];

    float acc_ae[8], acc_aa[8];
#pragma unroll
    for (int r = 0; r < 8; ++r) { acc_ae[r] = 0.f; acc_aa[r] = 0.f; }

    for (int jt = wave; jt < N / 16; jt += 8) {
        const int j0 = jt * 16;
        {   // agent-expert tile
            v8f c = wmma_xyT_tile(SA, ESA, i0, j0);
            const float yn = E2[j0 + nrow];
            const float vy = ew[j0 + nrow];
#pragma unroll
            for (int r = 0; r < 8; ++r) {
                float d2 = fmaxf(xm[r] + yn - 2.0f * c[r], 0.f) * (1.0f / DFEAT);
                float kk = expf(-g1 * d2) + expf(-g2 * d2);
                acc_ae[r] += kk * vy;
            }
        }
        {   // agent-agent tile
            v8f c = wmma_xyT_tile(SA, SA, i0, j0);
            const float yn = X2[j0 + nrow];
            const float vy = w[j0 + nrow];
#pragma unroll
            for (int r = 0; r < 8; ++r) {
                float d2 = fmaxf(xm[r] + yn - 2.0f * c[r], 0.f) * (1.0f / DFEAT);
                float kk = expf(-g1 * d2) + expf(-g2 * d2);
                acc_aa[r] += kk * vy;
            }
        }
    }

    // reduce over the 16 lanes sharing a row (width-16 xor shuffles)
#pragma unroll
    for (int r = 0; r < 8; ++r) {
        float va = acc_ae[r], vb = acc_aa[r];
        for (int off = 8; off >= 1; off >>= 1) {
            va += __shfl_xor(va, off, 16);
            vb += __shfl_xor(vb, off, 16);
        }
        if ((lane & 15) == 0) {
            part_ae[wave][r + 8 * half] = va;
            part_aa[wave][r + 8 * half] = vb;
        }
    }
    __syncthreads();

    if (threadIdx.x < 16) {
        const int m = threadIdx.x;
        float sae = 0.f, saa = 0.f;
        for (int wv = 0; wv < 8; ++wv) { sae += part_ae[wv][m]; saa += part_aa[wv][m]; }
        const float wn_i = w[i0 + m] / Sw;
        out[i0 + m] = wn_i * (sae / Sew - saa / Sw);
    }
}

// ---------------------------------------------------------------------------
extern "C" void kernel_launch(void* const* d_in, const int* in_sizes, int n_in,
                              void* d_out, int out_size, void* d_ws, size_t ws_size,
                              hipStream_t stream) {
    const float* state   = (const float*)d_in[0];
    const float* action  = (const float*)d_in[1];
    const float* estate  = (const float*)d_in[2];
    const float* eaction = (const float*)d_in[3];
    const float* w       = (const float*)d_in[4];
    const float* ew      = (const float*)d_in[5];
    float* out = (float*)d_out;

    if (ws_size < WS_TOTAL * sizeof(float)) return;
    float* ws   = (float*)d_ws;
    float* SA   = ws + WS_SA;
    float* ESA  = ws + WS_ESA;
    float* X2   = ws + WS_X2;
    float* E2   = ws + WS_E2;
    float* HIST = ws + WS_HIST;
    float* P    = ws + WS_P;

    k_zero<<<8, 256, 0, stream>>>(w, ew, HIST, P);
    k_prep<<<N / 256, 256, 0, stream>>>(state, action, estate, eaction, SA, ESA, X2, E2);

    // median of D_ae weighted by w (x) ew  ->  g1 = P[2]
    k_hist<<<N / 16, 256, 0, stream>>>(SA, ESA, X2, E2, w, ew, P, HIST);
    k_select<<<1, 256, 0, stream>>>(HIST, P, 0, 0);
    k_hist<<<N / 16, 256, 0, stream>>>(SA, ESA, X2, E2, w, ew, P, HIST);
    k_select<<<1, 256, 0, stream>>>(HIST, P, 1, 2);

    // median of D_ee weighted by ew (x) ew  ->  g2 = P[3]
    k_hist<<<N / 16, 256, 0, stream>>>(ESA, ESA, E2, E2, ew, ew, P, HIST);
    k_select<<<1, 256, 0, stream>>>(HIST, P, 0, 0);
    k_hist<<<N / 16, 256, 0, stream>>>(ESA, ESA, E2, E2, ew, ew, P, HIST);
    k_select<<<1, 256, 0, stream>>>(HIST, P, 1, 3);

    // fused two-bandwidth kernel row sums -> output
    k_out<<<N / 16, 256, 0, stream>>>(SA, ESA, X2, E2, w, ew, P, out);
    (void)in_sizes; (void)n_in; (void)out_size;
}